// BasePointSAModule_75685913690516
// MI455X (gfx1250) — compile-verified
//
#include <hip/hip_runtime.h>

#define B_   4
#define N_   16384
#define C_   128
#define M_   1024
#define K_   32
#define P_   (B_*M_*K_)      // 131072 = 2^17
#define R2_  0.16f           // RADIUS^2
#define EPS_ 1e-5f

typedef __attribute__((ext_vector_type(2))) float v2f;
typedef __attribute__((ext_vector_type(8))) float v8f;

// ---------------------------------------------------------------------------
// Kernel 1: gather new_xyz + ball query (wave32: one wave per query point).
// Replicates "first K indices within radius, in index order, pad with first".
// ---------------------------------------------------------------------------
__global__ void ballquery_kernel(const float* __restrict__ xyz,
                                 const int*   __restrict__ indices,
                                 float*       __restrict__ new_xyz,
                                 int*         __restrict__ idxbuf)
{
    int gwave = (blockIdx.x * blockDim.x + threadIdx.x) >> 5;
    int lane  = threadIdx.x & 31;
    if (gwave >= B_ * M_) return;

    int b = gwave / M_;
    const float* pts = xyz + (size_t)b * N_ * 3;
    int ctr = indices[gwave];
    float cx = pts[ctr*3+0], cy = pts[ctr*3+1], cz = pts[ctr*3+2];
    if (lane < 3) new_xyz[gwave*3 + lane] = pts[ctr*3 + lane];

    int* out = idxbuf + gwave * K_;
    int cnt = 0;
    int firstIdx = -1;

    for (int base = 0; base < N_; base += 32) {
        int i = base + lane;
        float dx = pts[i*3+0] - cx;
        float dy = pts[i*3+1] - cy;
        float dz = pts[i*3+2] - cz;
        bool within = (dx*dx + dy*dy + dz*dz) < R2_;
        unsigned mask = (unsigned)__ballot(within);       // wave32: low 32 bits
        if (firstIdx < 0 && mask) firstIdx = base + (__ffs(mask) - 1);
        if (within) {
            int slot = cnt + __popc(mask & ((1u << lane) - 1u));
            if (slot < K_) out[slot] = i;
        }
        cnt += __popc(mask);
        if (cnt >= K_) break;                             // uniform across wave
    }
    if (firstIdx < 0) firstIdx = 0;                       // argsort-of-equal-keys -> 0
    if (cnt > K_) cnt = K_;
    for (int s = cnt + lane; s < K_; s += 32) out[s] = firstIdx;
}

// ---------------------------------------------------------------------------
// Kernel 2: stage-1 GEMM with fused gather. Block = 8 waves; LDS holds the
// 132x16 X column tile (grouped_xyz ++ grouped_feat ++ zero pad). All A/B
// operands are preloaded into registers, then 33 V_WMMA_F32_16X16X4_F32 are
// issued back-to-back (D->C accumulation needs no NOPs per ISA 7.12.1).
// ---------------------------------------------------------------------------
__global__ void gemm1_kernel(const float* __restrict__ xyz,
                             const float* __restrict__ feat,
                             const int*   __restrict__ idxbuf,
                             const float* __restrict__ new_xyz,
                             const float* __restrict__ W,    // (128,131)
                             const float* __restrict__ bias, // (128)
                             float*       __restrict__ Y)    // (128,P)
{
    constexpr int CIN = 131, CINPAD = 132, NIT = CINPAD / 4;
    __shared__ float sX[CINPAD * 16];
    __shared__ int   sIdx[16];
    __shared__ int   sB[16];
    __shared__ float sCtr[16 * 3];

    int ptile = blockIdx.x * 16;
    int t = threadIdx.x;
    int wave = t >> 5, lane = t & 31;
    int col   = lane & 15;
    int khalf = (lane >> 4) * 2;           // lanes 16-31 hold K+2/K+3
    int rowA  = wave * 16 + col;           // A-matrix row (Cout channel)

    // Preload A (weights) into registers while LDS is being filled.
    v2f aR[NIT];
#pragma unroll
    for (int it = 0; it < NIT; ++it) {
        int ka = it * 4 + khalf;
        aR[it].x = (ka     < CIN) ? W[rowA * CIN + ka]     : 0.f;
        aR[it].y = (ka + 1 < CIN) ? W[rowA * CIN + ka + 1] : 0.f;
    }

    if (t < 16) {
        int p = ptile + t;
        int q = p >> 5;            // (b*M + m)
        int k = p & 31;
        sIdx[t] = idxbuf[q * K_ + k];
        sB[t]   = q / M_;
        sCtr[t*3+0] = new_xyz[q*3+0];
        sCtr[t*3+1] = new_xyz[q*3+1];
        sCtr[t*3+2] = new_xyz[q*3+2];
    }
    __syncthreads();

    for (int e = t; e < CINPAD * 16; e += 256) {
        int c = e >> 4, cc = e & 15;
        int i = sIdx[cc], b = sB[cc];
        float v;
        if (c < 3)        v = xyz[((size_t)b * N_ + i) * 3 + c] - sCtr[cc*3 + c];
        else if (c < CIN) v = feat[(size_t)b * C_ * N_ + (size_t)(c - 3) * N_ + i];
        else              v = 0.f;
        sX[c * 16 + cc] = v;
    }
    __syncthreads();

    // Preload B column operands, then dense WMMA chain.
    v2f bR[NIT];
#pragma unroll
    for (int it = 0; it < NIT; ++it) {
        int ka = it * 4 + khalf;
        bR[it].x = sX[ka * 16 + col];
        bR[it].y = sX[(ka + 1) * 16 + col];
    }
    v8f acc = {};
#pragma unroll
    for (int it = 0; it < NIT; ++it)
        acc = __builtin_amdgcn_wmma_f32_16x16x4_f32(
                  false, aR[it], false, bR[it], (short)0, acc, false, false);

    int moff = (lane >> 4) << 3;           // C/D: vgpr v -> row v / v+8
#pragma unroll
    for (int v = 0; v < 8; v++) {
        int r = wave * 16 + v + moff;
        Y[(size_t)r * P_ + ptile + col] = acc[v] + bias[r];
    }
}

// ---------------------------------------------------------------------------
// Generic GEMM (stages 2/3): X (CIN,P) x W (Cout,CIN) -> Y (Cout,P) + bias.
// gridDim.y slabs of 128 Cout channels each. Same register-preload pattern.
// ---------------------------------------------------------------------------
template<int CIN>
__global__ void gemm_kernel(const float* __restrict__ X,
                            const float* __restrict__ W,
                            const float* __restrict__ bias,
                            float*       __restrict__ Y)
{
    constexpr int NIT = CIN / 4;
    __shared__ float sX[CIN * 16];
    int ptile = blockIdx.x * 16;
    int t = threadIdx.x;
    int wave = t >> 5, lane = t & 31;
    int col   = lane & 15;
    int khalf = (lane >> 4) * 2;
    int coutBase = (blockIdx.y * 8 + wave) * 16;
    int rowA  = coutBase + col;

    v2f aR[NIT];
#pragma unroll
    for (int it = 0; it < NIT; ++it) {
        int ka = it * 4 + khalf;
        aR[it].x = W[rowA * CIN + ka];
        aR[it].y = W[rowA * CIN + ka + 1];
    }

    for (int e = t; e < CIN * 16; e += 256) {
        int c = e >> 4, cc = e & 15;
        sX[e] = X[(size_t)c * P_ + ptile + cc];
    }
    __syncthreads();

    v2f bR[NIT];
#pragma unroll
    for (int it = 0; it < NIT; ++it) {
        int ka = it * 4 + khalf;
        bR[it].x = sX[ka * 16 + col];
        bR[it].y = sX[(ka + 1) * 16 + col];
    }
    v8f acc = {};
#pragma unroll
    for (int it = 0; it < NIT; ++it)
        acc = __builtin_amdgcn_wmma_f32_16x16x4_f32(
                  false, aR[it], false, bR[it], (short)0, acc, false, false);

    int moff = (lane >> 4) << 3;
#pragma unroll
    for (int v = 0; v < 8; v++) {
        int r = coutBase + v + moff;
        Y[(size_t)r * P_ + ptile + col] = acc[v] + bias[r];
    }
}

// ---------------------------------------------------------------------------
// Per-channel batch statistics (mean / biased var) over P positions. b128.
// ---------------------------------------------------------------------------
__global__ void stats_kernel(const float* __restrict__ Y,
                             float* __restrict__ mean,
                             float* __restrict__ var)
{
    __shared__ float ss[256], ss2[256];
    int c = blockIdx.x;
    const float4* row = (const float4*)(Y + (size_t)c * P_);
    float s = 0.f, s2 = 0.f;
    for (int p = threadIdx.x; p < P_ / 4; p += 256) {
        float4 v = row[p];
        s  += v.x + v.y + v.z + v.w;
        s2 += v.x*v.x + v.y*v.y + v.z*v.z + v.w*v.w;
    }
    ss[threadIdx.x] = s; ss2[threadIdx.x] = s2;
    __syncthreads();
    for (int off = 128; off > 0; off >>= 1) {
        if (threadIdx.x < off) {
            ss[threadIdx.x]  += ss[threadIdx.x + off];
            ss2[threadIdx.x] += ss2[threadIdx.x + off];
        }
        __syncthreads();
    }
    if (threadIdx.x == 0) {
        float mu = ss[0] / (float)P_;
        mean[c] = mu;
        var[c]  = ss2[0] / (float)P_ - mu * mu;
    }
}

// ---------------------------------------------------------------------------
// BN + ReLU elementwise (stages 1/2 -> next GEMM input), float4 vectorized.
// ---------------------------------------------------------------------------
__global__ void bnrelu_kernel(const float4* __restrict__ Y,
                              const float*  __restrict__ gamma,
                              const float*  __restrict__ beta,
                              const float*  __restrict__ mean,
                              const float*  __restrict__ var,
                              float4*       __restrict__ Xn, int n4)
{
    int tid = blockIdx.x * 256 + threadIdx.x;
    if (tid >= n4) return;
    int c = tid >> 15;                       // P_/4 == 2^15
    float g = gamma[c], be = beta[c], mu = mean[c];
    float rs = rsqrtf(var[c] + EPS_);
    float4 v = Y[tid];
    v.x = g * (v.x - mu) * rs + be;  v.x = v.x > 0.f ? v.x : 0.f;
    v.y = g * (v.y - mu) * rs + be;  v.y = v.y > 0.f ? v.y : 0.f;
    v.z = g * (v.z - mu) * rs + be;  v.z = v.z > 0.f ? v.z : 0.f;
    v.w = g * (v.w - mu) * rs + be;  v.w = v.w > 0.f ? v.w : 0.f;
    Xn[tid] = v;
}

// ---------------------------------------------------------------------------
// Stage-3 BN + ReLU fused with max-pool over K and final write (B,256,M).
// ---------------------------------------------------------------------------
__global__ void final_kernel(const float* __restrict__ Y,
                             const float* __restrict__ gamma,
                             const float* __restrict__ beta,
                             const float* __restrict__ mean,
                             const float* __restrict__ var,
                             float*       __restrict__ out)
{
    int t = blockIdx.x * 256 + threadIdx.x;     // t = (b*256 + c)*M + m
    if (t >= B_ * 256 * M_) return;
    int m = t & (M_ - 1);
    int c = (t >> 10) & 255;
    int b = t >> 18;
    float g = gamma[c], be = beta[c], mu = mean[c];
    float rs = rsqrtf(var[c] + EPS_);
    const float4* row = (const float4*)(Y + (size_t)c * P_ + ((size_t)(b * M_ + m) << 5));
    float best = 0.f;                            // post-ReLU values are >= 0
#pragma unroll
    for (int k = 0; k < K_ / 4; k++) {
        float4 v4 = row[k];
        float v;
        v = g * (v4.x - mu) * rs + be; v = v > 0.f ? v : 0.f; best = v > best ? v : best;
        v = g * (v4.y - mu) * rs + be; v = v > 0.f ? v : 0.f; best = v > best ? v : best;
        v = g * (v4.z - mu) * rs + be; v = v > 0.f ? v : 0.f; best = v > best ? v : best;
        v = g * (v4.w - mu) * rs + be; v = v > 0.f ? v : 0.f; best = v > best ? v : best;
    }
    out[t] = best;
}

// ---------------------------------------------------------------------------
extern "C" void kernel_launch(void* const* d_in, const int* in_sizes, int n_in,
                              void* d_out, int out_size, void* d_ws, size_t ws_size,
                              hipStream_t stream)
{
    const float* xyz     = (const float*)d_in[0];
    const float* feat    = (const float*)d_in[1];
    const int*   indices = (const int*)  d_in[2];
    const float* w1 = (const float*)d_in[3];
    const float* b1 = (const float*)d_in[4];
    const float* g1 = (const float*)d_in[5];
    const float* be1= (const float*)d_in[6];
    const float* w2 = (const float*)d_in[7];
    const float* b2 = (const float*)d_in[8];
    const float* g2 = (const float*)d_in[9];
    const float* be2= (const float*)d_in[10];
    const float* w3 = (const float*)d_in[11];
    const float* b3 = (const float*)d_in[12];
    const float* g3 = (const float*)d_in[13];
    const float* be3= (const float*)d_in[14];

    float* out = (float*)d_out;
    float* new_xyz = out;                        // (B,M,3) = 12288 floats
    float* featout = out + B_ * M_ * 3;          // (B,256,M)

    char* ws = (char*)d_ws;
    int*   idxbuf = (int*)ws;                                      // B*M*K ints
    float* bufA = (float*)(ws + (size_t)(B_*M_*K_) * sizeof(int)); // 128*P f32
    float* bufB = bufA + (size_t)128 * P_;                         // 256*P f32
    float* mean = bufB + (size_t)256 * P_;
    float* var  = mean + 256;

    // 1) new_xyz gather + ball query
    ballquery_kernel<<<(B_*M_)/8, 256, 0, stream>>>(xyz, indices, new_xyz, idxbuf);

    // 2) stage 1: fused gather + GEMM (131->128)
    gemm1_kernel<<<P_/16, 256, 0, stream>>>(xyz, feat, idxbuf, new_xyz, w1, b1, bufB);
    stats_kernel<<<128, 256, 0, stream>>>(bufB, mean, var);
    bnrelu_kernel<<<(128*P_/4)/256, 256, 0, stream>>>((const float4*)bufB, g1, be1, mean, var,
                                                      (float4*)bufA, 128*P_/4);

    // 3) stage 2: GEMM (128->128)
    gemm_kernel<128><<<dim3(P_/16, 1), 256, 0, stream>>>(bufA, w2, b2, bufB);
    stats_kernel<<<128, 256, 0, stream>>>(bufB, mean, var);
    bnrelu_kernel<<<(128*P_/4)/256, 256, 0, stream>>>((const float4*)bufB, g2, be2, mean, var,
                                                      (float4*)bufA, 128*P_/4);

    // 4) stage 3: GEMM (128->256), then fused BN+ReLU+maxpool
    gemm_kernel<128><<<dim3(P_/16, 2), 256, 0, stream>>>(bufA, w3, b3, bufB);
    stats_kernel<<<256, 256, 0, stream>>>(bufB, mean, var);
    final_kernel<<<(B_*256*M_)/256, 256, 0, stream>>>(bufB, g3, be3, mean, var, featout);
}